// Router_53077205843992
// MI455X (gfx1250) — compile-verified
//
#include <hip/hip_runtime.h>

#define N_TOK 4096
#define DIM   1024
#define NEXP  8
#define TOPK  2
#define CAP   1280   // int(4096*2*1.25/8)

typedef __attribute__((ext_vector_type(2))) float v2f;
typedef __attribute__((ext_vector_type(8))) float v8f;

// ---------------------------------------------------------------------------
// K1: logits[t][e] = dot(x[t], W_gate[e]) via V_WMMA_F32_16X16X4_F32.
// One wave handles a 16-token tile; K loop of 256 steps (K=4 each).
// A (16x4, MxK): lanes 0-15 -> M=lane, K={0,1} in the 2 VGPRs;
//                lanes 16-31 -> M=lane-16, K={2,3}.
// B (4x16, KxN): same lane-half split for K pairs; col N = lane%16.
// C (16x16):     VGPR v -> row v (lanes 0-15) / row v+8 (lanes 16-31), col=lane%16.
// Cols 8..15 of B are garbage (clamped W row) -> only affects unused C cols.
// ---------------------------------------------------------------------------
__global__ __launch_bounds__(256) void moe_logits_wmma(
    const float* __restrict__ x, const float* __restrict__ wg,
    float* __restrict__ logits)
{
    const int wave = (blockIdx.x * blockDim.x + threadIdx.x) >> 5;
    const int lane = threadIdx.x & 31;
    const int tok0 = wave * 16;
    const int m    = lane & 15;          // A row / C col index base
    const int koff = (lane < 16) ? 0 : 2;

    const float* xrow = x  + (size_t)(tok0 + m) * DIM + koff;
    const float* wrow = wg + (size_t)(m & (NEXP - 1)) * DIM + koff; // clamp pad cols

    v8f c = {};
    for (int k = 0; k < DIM; k += 4) {
        v2f a = *(const v2f*)(xrow + k);
        v2f b = *(const v2f*)(wrow + k);
        c = __builtin_amdgcn_wmma_f32_16x16x4_f32(
                /*neg_a=*/false, a, /*neg_b=*/false, b,
                /*c_mod=*/(short)0, c, /*reuse_a=*/false, /*reuse_b=*/false);
    }

    if (m < NEXP) {   // only cols 0..7 are real experts
        const int mbase = (lane < 16) ? 0 : 8;
#pragma unroll
        for (int v = 0; v < 8; ++v)
            logits[(size_t)(tok0 + mbase + v) * NEXP + m] = c[v];
    }
}

// ---------------------------------------------------------------------------
// K2: per-token top-2 (stable, lower index wins ties) + softmax over the pair.
// ---------------------------------------------------------------------------
__global__ __launch_bounds__(256) void moe_topk(
    const float* __restrict__ logits, float* __restrict__ gates_out,
    int* __restrict__ e_flat, float* __restrict__ g_flat)
{
    int t = blockIdx.x * blockDim.x + threadIdx.x;
    if (t >= N_TOK) return;
    float l[NEXP];
#pragma unroll
    for (int e = 0; e < NEXP; ++e) l[e] = logits[(size_t)t * NEXP + e];

    int i0 = 0; float v0 = l[0];
#pragma unroll
    for (int e = 1; e < NEXP; ++e) if (l[e] > v0) { v0 = l[e]; i0 = e; }
    int i1 = (i0 == 0) ? 1 : 0; float v1 = l[i1];
#pragma unroll
    for (int e = 0; e < NEXP; ++e)
        if (e != i0 && e != i1 && l[e] > v1) { v1 = l[e]; i1 = e; }

    float ex = __expf(v1 - v0);          // v0 >= v1, numerically safe
    float g0 = 1.0f / (1.0f + ex);
    float g1 = ex * g0;

    gates_out[t * 2 + 0] = g0;  gates_out[t * 2 + 1] = g1;
    e_flat[t * 2 + 0] = i0;     e_flat[t * 2 + 1] = i1;
    g_flat[t * 2 + 0] = g0;     g_flat[t * 2 + 1] = g1;
}

// ---------------------------------------------------------------------------
// K3: exact serial position assignment. 8192 slots, 1 workgroup of 1024
// threads, 8 slots/thread. Block-wide Hillis-Steele scan of 8-wide expert
// count vectors in LDS, then serial replay of each thread's 8 slots.
// ---------------------------------------------------------------------------
__global__ __launch_bounds__(1024) void moe_scan(
    const int* __restrict__ e_flat, int* __restrict__ pos)
{
    __shared__ int sc[1024][NEXP];
    const int t = threadIdx.x;

    int e_loc[8];
    int cnt[NEXP] = {0, 0, 0, 0, 0, 0, 0, 0};
#pragma unroll
    for (int i = 0; i < 8; ++i) {
        int e = e_flat[t * 8 + i];
        e_loc[i] = e;
        cnt[e]++;
    }
#pragma unroll
    for (int j = 0; j < NEXP; ++j) sc[t][j] = cnt[j];
    __syncthreads();

    for (int off = 1; off < 1024; off <<= 1) {
        int tmp[NEXP];
        if (t >= off) {
#pragma unroll
            for (int j = 0; j < NEXP; ++j) tmp[j] = sc[t - off][j];
        }
        __syncthreads();
        if (t >= off) {
#pragma unroll
            for (int j = 0; j < NEXP; ++j) sc[t][j] += tmp[j];
        }
        __syncthreads();
    }

    int run[NEXP];
#pragma unroll
    for (int j = 0; j < NEXP; ++j) run[j] = (t == 0) ? 0 : sc[t - 1][j];
#pragma unroll
    for (int i = 0; i < 8; ++i) {
        int e = e_loc[i];
        pos[t * 8 + i] = run[e]++;
    }
}

// ---------------------------------------------------------------------------
// K4: scatter. One block per slot: lane 0 writes the gate into exp_mask,
// 256 threads copy the 1024-float x row (float4) into exp_batches.
// ---------------------------------------------------------------------------
__global__ __launch_bounds__(256) void moe_scatter(
    const float* __restrict__ x, const int* __restrict__ e_flat,
    const int* __restrict__ pos, const float* __restrict__ g_flat,
    float* __restrict__ exp_mask, float* __restrict__ exp_batches)
{
    const int s = blockIdx.x;
    const int p = pos[s];
    if (p >= CAP) return;                     // dropped token-slot
    const int e   = e_flat[s];
    const int tok = s >> 1;

    if (threadIdx.x == 0)
        exp_mask[((size_t)tok * NEXP + e) * CAP + p] = g_flat[s];

    const float4* src = (const float4*)(x + (size_t)tok * DIM);
    float4*       dst = (float4*)(exp_batches + ((size_t)e * CAP + p) * DIM);
    dst[threadIdx.x] = src[threadIdx.x];      // 256 * 16B = 4KB row
}

// ---------------------------------------------------------------------------
extern "C" void kernel_launch(void* const* d_in, const int* in_sizes, int n_in,
                              void* d_out, int out_size, void* d_ws, size_t ws_size,
                              hipStream_t stream)
{
    const float* x  = (const float*)d_in[0];   // [2,2048,1024] f32
    const float* wg = (const float*)d_in[1];   // [8,1024] f32

    float* out         = (float*)d_out;
    float* gates       = out;                                        // 4096*2
    float* exp_mask    = out + (size_t)N_TOK * TOPK;                 // 4096*8*1280
    float* exp_batches = exp_mask + (size_t)N_TOK * NEXP * CAP;      // 8*1280*1024

    float* ws_logits = (float*)d_ws;                                 // 4096*8 f32
    int*   ws_e      = (int*)(ws_logits + (size_t)N_TOK * NEXP);     // 8192 i32
    float* ws_g      = (float*)(ws_e + N_TOK * TOPK);                // 8192 f32
    int*   ws_pos    = (int*)(ws_g + N_TOK * TOPK);                  // 8192 i32

    // Zero-fill outputs (exp_mask/exp_batches are sparse-scattered; ~210 MB,
    // the bandwidth-bound bulk of this op).
    hipMemsetAsync(d_out, 0, (size_t)out_size * sizeof(float), stream);

    // 256 tiles of 16 tokens -> 256 waves -> 32 blocks x 8 waves.
    moe_logits_wmma<<<32, 256, 0, stream>>>(x, wg, ws_logits);
    moe_topk<<<(N_TOK + 255) / 256, 256, 0, stream>>>(ws_logits, gates, ws_e, ws_g);
    moe_scan<<<1, 1024, 0, stream>>>(ws_e, ws_pos);
    moe_scatter<<<N_TOK * TOPK, 256, 0, stream>>>(x, ws_e, ws_pos, ws_g,
                                                  exp_mask, exp_batches);
}